// JINOLoss_90134183674123
// MI455X (gfx1250) — compile-verified
//
#include <hip/hip_runtime.h>

typedef __attribute__((ext_vector_type(2))) float v2f;
typedef __attribute__((ext_vector_type(8))) float v8f;

#define TEMP_INV (1.0f / 0.07f)

// One block per output row (b, h).  Computes the banded column correlation
// Dcol(h', x, d) = sum_c ft[b,c,h',x] * fs[b,c,h',x-d]  (d in [0,20]) with
// fp32 WMMA 16x16x4 tiles, then 4x4 box-sums + band log-softmax CE in LDS.
__global__ __launch_bounds__(256) void jino_band_loss_kernel(
    const float* __restrict__ fs, const float* __restrict__ ft,
    float* __restrict__ partial)
{
  constexpr int C = 48, Hf = 99, Wf = 195;
  constexpr int Ws = 192;          // output columns (Wf - 4 + 1)
  constexpr int ND = 21;           // band width: d = i - j in [0, 20]
  constexpr int HW = Hf * Wf;
  constexpr int NX = Wf;           // 195 column positions

  const int h = blockIdx.x;        // 0..95
  const int b = blockIdx.y;        // 0..3

  extern __shared__ float smem[];
  float* Dl  = smem;                     // [4][NX][ND] banded correlations
  float* nT  = Dl + 4 * NX * ND;         // [4][NX] column |t|^2
  float* nS  = nT + 4 * NX;              // [4][NX] column |s|^2
  float* red = nS + 4 * NX;              // [256] reduction

  const int tid  = threadIdx.x;
  const int lane = tid & 31;
  const int wave = tid >> 5;             // 8 waves

  const float* __restrict__ ftb = ft + (size_t)b * C * HW;
  const float* __restrict__ fsb = fs + (size_t)b * C * HW;

  // ---- per-column squared norms (plain VALU, trivial work) ----
  for (int job = tid; job < 4 * NX; job += 256) {
    const int ki = job / NX, x = job % NX;
    const float* pt = ftb + (h + ki) * Wf + x;
    const float* ps = fsb + (h + ki) * Wf + x;
    float st = 0.f, ss = 0.f;
    for (int c = 0; c < C; ++c) {
      float a = pt[c * HW]; st = fmaf(a, a, st);
      float e = ps[c * HW]; ss = fmaf(e, e, ss);
    }
    nT[job] = st;
    nS[job] = ss;
  }

  // ---- banded column correlation via V_WMMA_F32_16X16X4_F32 ----
  // Jobs: ki in [0,4) x 13 i-tiles x 3 j-tiles = 156 tiles, round-robin on waves.
  const int m  = lane & 15;              // M (and N) index within tile
  const int kb = (lane >> 4) << 1;       // K base: lanes 16-31 hold K=2,3
  const int hi = lane >> 4;
  for (int job = wave; job < 4 * 13 * 3; job += 8) {
    const int ki = job / 39;
    const int r  = job % 39;
    const int i0 = (r / 3) * 16;
    const int j0 = i0 - 32 + (r % 3) * 16;
    if (j0 + 15 < 0) continue;           // wave-uniform: tile entirely at x<0
    const int hh = h + ki;               // 0..98
    int ic = i0 + m; if (ic > NX - 1) ic = NX - 1;          // clamp, mask later
    int jc = j0 + m; if (jc < 0) jc = 0; if (jc > NX - 1) jc = NX - 1;
    const float* At = ftb + hh * Wf + ic;                    // A: T[i', c]
    const float* Bt = fsb + hh * Wf + jc;                    // B: S[j', c]
    v8f acc = {};
    for (int c0 = 0; c0 < C; c0 += 4) {
      v2f a, bb;
      a.x  = At[(c0 + kb) * HW];
      a.y  = At[(c0 + kb + 1) * HW];
      bb.x = Bt[(c0 + kb) * HW];
      bb.y = Bt[(c0 + kb + 1) * HW];
      acc = __builtin_amdgcn_wmma_f32_16x16x4_f32(
          false, a, false, bb, (short)0, acc, false, false);
    }
    // scatter in-band results into LDS (each (ki, i', d) owned by one tile)
    for (int v = 0; v < 8; ++v) {
      const int ii = i0 + v + (hi << 3);
      const int jj = j0 + m;
      const int d  = ii - jj;
      if (ii <= NX - 1 && jj >= 0 && d >= 0 && d < ND)
        Dl[(ki * NX + ii) * ND + d] = acc[v];
    }
  }
  __syncthreads();

  // ---- 4x4 box sums + band softmax cross-entropy (one i per thread) ----
  float loss = 0.f;
  if (tid < Ws) {
    const int i = tid;
    float ntt = 0.f;
    for (int ki = 0; ki < 4; ++ki)
      for (int kj = 0; kj < 4; ++kj)
        ntt += nT[ki * NX + i + kj];
    const float lt = fmaxf(sqrtf(ntt), 1e-12f);
    const int dmax = (i < ND - 1) ? i : ND - 1;
    float mx = -3.0e38f, se = 0.f, logit0 = 0.f;
    for (int d = 0; d <= dmax; ++d) {       // online logsumexp over the band
      const int j = i - d;
      float num = 0.f, nss = 0.f;
      for (int ki = 0; ki < 4; ++ki)
        for (int kj = 0; kj < 4; ++kj) {
          num += Dl[(ki * NX + i + kj) * ND + d];
          nss += nS[ki * NX + j + kj];
        }
      const float ls = fmaxf(sqrtf(nss), 1e-12f);
      const float lg = num / (lt * ls) * TEMP_INV;
      if (d == 0) logit0 = lg;
      if (lg > mx) { se *= expf(mx - lg); mx = lg; }
      se += expf(lg - mx);
    }
    loss = (mx + logf(se)) - logit0;        // -log p(diagonal)
  }
  red[tid] = loss;
  __syncthreads();
  for (int s = 128; s > 0; s >>= 1) {
    if (tid < s) red[tid] += red[tid + s];
    __syncthreads();
  }
  if (tid == 0) partial[b * gridDim.x + h] = red[0];
}

// Deterministic final reduction: 384 partials -> scalar mean.
__global__ __launch_bounds__(512) void jino_reduce_kernel(
    const float* __restrict__ partial, float* __restrict__ out, int n, float inv_cnt)
{
  __shared__ float red[512];
  const int tid = threadIdx.x;
  red[tid] = (tid < n) ? partial[tid] : 0.f;
  __syncthreads();
  for (int s = 256; s > 0; s >>= 1) {
    if (tid < s) red[tid] += red[tid + s];
    __syncthreads();
  }
  if (tid == 0) out[0] = red[0] * inv_cnt;
}

extern "C" void kernel_launch(void* const* d_in, const int* in_sizes, int n_in,
                              void* d_out, int out_size, void* d_ws, size_t ws_size,
                              hipStream_t stream) {
  (void)in_sizes; (void)n_in; (void)out_size; (void)ws_size;
  const float* fs = (const float*)d_in[0];
  const float* ft = (const float*)d_in[1];
  // d_in[2] = ps (4), d_in[3] = stride (4): baked into the kernel constants.
  float* partial = (float*)d_ws;            // 384 floats
  float* out = (float*)d_out;

  constexpr int Hs = 96, B = 4, Ws = 192;
  constexpr int NX = 195, ND = 21;
  const size_t lds_bytes = (size_t)(4 * NX * ND + 2 * 4 * NX + 256) * sizeof(float);

  jino_band_loss_kernel<<<dim3(Hs, B), 256, lds_bytes, stream>>>(fs, ft, partial);
  jino_reduce_kernel<<<1, 512, 0, stream>>>(partial, out, B * Hs,
                                            1.0f / (float)(B * Hs * Ws));
}